// MLPAttention_78039555768791
// MI455X (gfx1250) — compile-verified
//
#include <hip/hip_runtime.h>
#include <hip/hip_bf16.h>
#include <math.h>

typedef __attribute__((ext_vector_type(2))) float v2f;
typedef __attribute__((ext_vector_type(8))) float v8f;

#define BB 4
#define LL 512
#define DD 128
// M = BB*LL = 2048 rows for the projection GEMMs

// ---- fast tanh: prefer the gfx1250 V_TANH_F32 builtin if this clang has it ----
__device__ __forceinline__ float fast_tanh(float x) {
#if defined(__has_builtin)
#if __has_builtin(__builtin_amdgcn_tanhf)
    return __builtin_amdgcn_tanhf(x);
#elif __has_builtin(__builtin_amdgcn_tanh_f32)
    return __builtin_amdgcn_tanh_f32(x);
#else
    return tanhf(x);
#endif
#else
    return tanhf(x);
#endif
}

// ============================================================================
// Kernel A: q = inp @ Wq^T + bq ; k = inp @ Wk^T   via V_WMMA_F32_16X16X4_F32
// One wave32 owns one 16x16 output tile of BOTH q and k (shared A loads).
// Grid: 256 blocks x 128 threads = 1024 waves = (2048/16) x (128/16) tiles.
//
// A-frag (16x4 f32, 2 VGPRs): lane&15 = M row; lane>>4 selects K-pair
//   -> each lane loads inp[m, k0+2*(lane>>4) .. +1]  (contiguous 8B)
// B-frag (4x16 f32, 2 VGPRs): lane&15 = N col; same K-pair select
//   -> each lane loads W[n, k0+2*(lane>>4) .. +1]    (W^T as KxN, contiguous 8B)
// C/D (16x16 f32, 8 VGPRs): M = r + (lane>>4)*8, N = lane&15
// ============================================================================
__global__ __launch_bounds__(128)
void proj_qk_wmma(const float* __restrict__ inp,
                  const float* __restrict__ Wq, const float* __restrict__ bq,
                  const float* __restrict__ Wk,
                  float* __restrict__ qo, float* __restrict__ ko)
{
    const int wave = threadIdx.x >> 5;
    const int lane = threadIdx.x & 31;
    const int tile = blockIdx.x * 4 + wave;       // 0..1023
    const int m0   = (tile >> 3) << 4;            // 0..2032 step 16
    const int n0   = (tile & 7) << 4;             // 0..112  step 16
    const int r    = lane & 15;
    const int koff = (lane >> 4) << 1;            // 0 or 2

    const float* arow  = inp + (size_t)(m0 + r) * DD + koff;
    const float* bqrow = Wq  + (size_t)(n0 + r) * DD + koff;
    const float* bkrow = Wk  + (size_t)(n0 + r) * DD + koff;

    v8f cq = {};
    v8f ck = {};
    for (int kk = 0; kk < DD; kk += 4) {
        v2f a   = *(const v2f*)(arow  + kk);
        v2f bq_ = *(const v2f*)(bqrow + kk);
        v2f bk_ = *(const v2f*)(bkrow + kk);
        // (neg_a, A, neg_b, B, c_mod, C, reuse_a, reuse_b)
        cq = __builtin_amdgcn_wmma_f32_16x16x4_f32(false, a, false, bq_,
                                                   (short)0, cq, false, false);
        ck = __builtin_amdgcn_wmma_f32_16x16x4_f32(false, a, false, bk_,
                                                   (short)0, ck, false, false);
    }

    const int   nn    = n0 + (lane & 15);
    const float bias  = bq[nn];
    const int   mbase = m0 + ((lane >> 4) << 3);
#pragma unroll
    for (int rr = 0; rr < 8; ++rr) {
        const int m = mbase + rr;
        qo[(size_t)m * DD + nn] = cq[rr] + bias;
        ko[(size_t)m * DD + nn] = ck[rr];
    }
}

// ============================================================================
// Kernel B: score[b,i,j] = sum_d tanh(q[b,j,d] + k[b,i,d]) * v[d], then
// softmax over j. One workgroup (256 threads = 8 wave32) per (b,i) row.
// k-row and v staged in LDS; lanes split across d (coalesced 128B q reads);
// dot reduced with 5 __shfl_xor steps (wave32).
// ============================================================================
__global__ __launch_bounds__(256)
void score_softmax(const float* __restrict__ q, const float* __restrict__ k,
                   const float* __restrict__ v, float* __restrict__ out)
{
    const int bi = blockIdx.x;          // b*LL + i
    const int b  = bi >> 9;             // /512

    __shared__ float s_k[DD];
    __shared__ float s_v[DD];
    __shared__ float s_score[LL];
    __shared__ float s_red[8];

    const int tid  = threadIdx.x;
    const int lane = tid & 31;
    const int wave = tid >> 5;          // 0..7

    if (tid < DD) {
        s_k[tid] = k[(size_t)bi * DD + tid];
        s_v[tid] = v[tid];
    }
    __syncthreads();

    // per-lane slices of k-row and v (d = lane, lane+32, lane+64, lane+96)
    const float kv0 = s_k[lane +  0], kv1 = s_k[lane + 32];
    const float kv2 = s_k[lane + 64], kv3 = s_k[lane + 96];
    const float vv0 = s_v[lane +  0], vv1 = s_v[lane + 32];
    const float vv2 = s_v[lane + 64], vv3 = s_v[lane + 96];

    const float* qb = q + (size_t)b * LL * DD;

    // each wave covers 64 j values
    for (int j = wave; j < LL; j += 8) {
        const float* qr = qb + (size_t)j * DD + lane;
        float acc = fast_tanh(qr[ 0] + kv0) * vv0
                  + fast_tanh(qr[32] + kv1) * vv1
                  + fast_tanh(qr[64] + kv2) * vv2
                  + fast_tanh(qr[96] + kv3) * vv3;
#pragma unroll
        for (int off = 16; off > 0; off >>= 1)
            acc += __shfl_xor(acc, off, 32);
        if (lane == 0) s_score[j] = acc;
    }
    __syncthreads();

    // ---- softmax over the 512 scores (each thread owns j=tid and j=tid+256) ----
    const float a0 = s_score[tid];
    const float a1 = s_score[tid + 256];

    float m = fmaxf(a0, a1);
#pragma unroll
    for (int off = 16; off > 0; off >>= 1)
        m = fmaxf(m, __shfl_xor(m, off, 32));
    if (lane == 0) s_red[wave] = m;
    __syncthreads();

    float bm = s_red[0];
#pragma unroll
    for (int w = 1; w < 8; ++w) bm = fmaxf(bm, s_red[w]);

    const float e0 = __expf(a0 - bm);
    const float e1 = __expf(a1 - bm);

    float s = e0 + e1;
#pragma unroll
    for (int off = 16; off > 0; off >>= 1)
        s += __shfl_xor(s, off, 32);
    __syncthreads();                    // all reads of s_red (max) done
    if (lane == 0) s_red[wave] = s;
    __syncthreads();

    float bs = s_red[0];
#pragma unroll
    for (int w = 1; w < 8; ++w) bs += s_red[w];
    const float inv = 1.0f / bs;

    float* orow = out + (size_t)bi * LL;
    orow[tid]       = e0 * inv;
    orow[tid + 256] = e1 * inv;
}

// ============================================================================
// Launcher. Workspace layout: q (2048*128 f32, 1MB) | k (2048*128 f32, 1MB).
// ============================================================================
extern "C" void kernel_launch(void* const* d_in, const int* in_sizes, int n_in,
                              void* d_out, int out_size, void* d_ws, size_t ws_size,
                              hipStream_t stream) {
    (void)in_sizes; (void)n_in; (void)out_size; (void)ws_size;
    const float* inp = (const float*)d_in[0];
    const float* Wq  = (const float*)d_in[1];
    const float* bq  = (const float*)d_in[2];
    const float* Wk  = (const float*)d_in[3];
    const float* v   = (const float*)d_in[4];
    float* out = (float*)d_out;

    float* qws = (float*)d_ws;                       // [2048,128]
    float* kws = qws + (size_t)BB * LL * DD;         // [2048,128]

    // Projections: 1024 tiles, 4 waves/block
    proj_qk_wmma<<<256, 128, 0, stream>>>(inp, Wq, bq, Wk, qws, kws);
    // Score + softmax: one block per (b,i)
    score_softmax<<<BB * LL, 256, 0, stream>>>(qws, kws, v, out);
}